// STN_88527865905290
// MI455X (gfx1250) — compile-verified
//
#include <hip/hip_runtime.h>

// ---------------------------------------------------------------------------
// CDNA5 / gfx1250 implementation of the point-cloud STN + KNN-feature network.
// All 1x1-conv GEMMs (fc2/fc3 are scalar) run through V_WMMA_F32_16X16X4_F32 (full f32
// precision, matching the f32 reference). KNN is a tiled-LDS scan with a
// per-thread top-32 replacement list (order-invariant downstream).
// ---------------------------------------------------------------------------

typedef float v2f __attribute__((ext_vector_type(2)));
typedef float v8f __attribute__((ext_vector_type(8)));

#define BN 8
#define NP 8192
#define MP 2048
#define KNN 32
#define GFDC 512
#define CI771 771           // 3 + 128 + 640
#define MC 128              // M-chunk for the conv_res-over-(K,M) stage
#define CCOLS (MC * KNN)    // 4096 columns per batch per chunk

__device__ __forceinline__ float lrelu_f(float x) { return x >= 0.0f ? x : 0.2f * x; }

// ---------------- WMMA GEMM: Y[b] = act(W * X[b] + bias (+ addin[b])) -------
// W: (Co, Ci) row-major. X: per-batch (Ci, cols) row-major, batch stride
// Ci*cols. Y/addin: (Co, cols), batch stride Co*cols. cols % 64 == 0.
// One wave computes a 16(Co) x 64(cols) tile: 4 f32 accumulators, A reused
// across the 4 column sub-tiles, Ci consumed in chunks of 4.
// The main loop is branch-free AND software-pipelined: chunk k+1's operands
// (A pair + 4 B pairs) are loaded while chunk k's four WMMAs execute, so the
// s_wait_loadcnt before the WMMA group covers loads issued a full chunk
// earlier. OOB Co rows (Co=3 layer) are handled by clamping the W row
// pointer in-bounds and scaling A by 0; Ci%4 is one guarded tail chunk.
// A layout (16x4 f32): lanes 0-15 hold K=0,1; lanes 16-31 hold K=2,3.
// B layout (4x16 f32): vgpr0 = rows K=0(/2), vgpr1 = rows K=1(/3).
// C/D layout (16x16 f32): vgpr r -> row r (lanes 0-15) / r+8 (lanes 16-31).
__global__ __launch_bounds__(256) void gemm_wmma_f32(
    const float* __restrict__ W, const float* __restrict__ bias,
    const float* __restrict__ X, float* __restrict__ Y,
    const float* __restrict__ addin,
    int Co, int Ci, int cols, int act)
{
    const int b = blockIdx.y;
    const float* Xb = X + (size_t)b * Ci * cols;
    float* Yb = Y + (size_t)b * Co * cols;
    const float* Ab = addin ? addin + (size_t)b * Co * cols : nullptr;

    const int lane = threadIdx.x & 31;
    const int wid  = threadIdx.x >> 5;
    const int colTiles = cols >> 6;
    const int coTiles  = (Co + 15) >> 4;
    const int tile = blockIdx.x * (blockDim.x >> 5) + wid;
    if (tile >= colTiles * coTiles) return;
    const int coT  = tile / colTiles;
    const int colT = tile - coT * colTiles;
    const int co0 = coT << 4;
    const int n0  = colT << 6;

    v8f acc[4] = {v8f{}, v8f{}, v8f{}, v8f{}};

    const int mrow  = lane & 15;
    const int khalf = (lane >> 4) << 1;       // 0 (lanes 0-15) or 2 (lanes 16-31)
    const int co    = co0 + mrow;
    const bool coOK = co < Co;
    const float asc = coOK ? 1.0f : 0.0f;     // zero A rows beyond Co (branch-free)
    const int cosafe = coOK ? co : (Co - 1);  // keep loads in-bounds
    const int nl    = n0 + (lane & 15);

    const float* __restrict__ wrow = W + (size_t)cosafe * Ci + khalf;
    const float* __restrict__ xk   = Xb + (size_t)khalf * cols + nl;
    const size_t xstep = (size_t)4 * cols;

    const int Ci4 = Ci & ~3;
    if (Ci4 > 0) {
        v2f a_cur, b0, b1, b2, b3;
        // ---- preload chunk 0 ----
        a_cur.x = wrow[0] * asc;
        a_cur.y = wrow[1] * asc;
        {
            const float* r0 = xk;
            const float* r1 = xk + cols;
            b0.x = r0[0];  b0.y = r1[0];
            b1.x = r0[16]; b1.y = r1[16];
            b2.x = r0[32]; b2.y = r1[32];
            b3.x = r0[48]; b3.y = r1[48];
        }
        // ---- steady state: load chunk kk while computing chunk kk-4 ----
        for (int kk = 4; kk < Ci4; kk += 4) {
            xk += xstep;
            v2f a_nxt, n0v, n1v, n2v, n3v;
            a_nxt.x = wrow[kk] * asc;
            a_nxt.y = wrow[kk + 1] * asc;
            const float* r0 = xk;
            const float* r1 = xk + cols;
            n0v.x = r0[0];  n0v.y = r1[0];
            n1v.x = r0[16]; n1v.y = r1[16];
            n2v.x = r0[32]; n2v.y = r1[32];
            n3v.x = r0[48]; n3v.y = r1[48];
            acc[0] = __builtin_amdgcn_wmma_f32_16x16x4_f32(false, a_cur, false, b0, (short)0, acc[0], false, false);
            acc[1] = __builtin_amdgcn_wmma_f32_16x16x4_f32(false, a_cur, false, b1, (short)0, acc[1], false, false);
            acc[2] = __builtin_amdgcn_wmma_f32_16x16x4_f32(false, a_cur, false, b2, (short)0, acc[2], false, false);
            acc[3] = __builtin_amdgcn_wmma_f32_16x16x4_f32(false, a_cur, false, b3, (short)0, acc[3], false, false);
            a_cur = a_nxt; b0 = n0v; b1 = n1v; b2 = n2v; b3 = n3v;
        }
        // ---- drain last full chunk ----
        acc[0] = __builtin_amdgcn_wmma_f32_16x16x4_f32(false, a_cur, false, b0, (short)0, acc[0], false, false);
        acc[1] = __builtin_amdgcn_wmma_f32_16x16x4_f32(false, a_cur, false, b1, (short)0, acc[1], false, false);
        acc[2] = __builtin_amdgcn_wmma_f32_16x16x4_f32(false, a_cur, false, b2, (short)0, acc[2], false, false);
        acc[3] = __builtin_amdgcn_wmma_f32_16x16x4_f32(false, a_cur, false, b3, (short)0, acc[3], false, false);
        xk += xstep;
    }
    if (Ci4 < Ci) {  // single guarded tail chunk (Ci % 4 != 0)
        const int ka = Ci4 + khalf;
        const bool k0ok = ka < Ci;
        const bool k1ok = (ka + 1) < Ci;
        v2f a;
        a.x = k0ok ? wrow[Ci4] * asc : 0.0f;
        a.y = k1ok ? wrow[Ci4 + 1] * asc : 0.0f;
        const float* r0 = xk;
        const float* r1 = xk + cols;
#pragma unroll
        for (int s = 0; s < 4; ++s) {
            v2f bv;
            bv.x = k0ok ? r0[s * 16] : 0.0f;
            bv.y = k1ok ? r1[s * 16] : 0.0f;
            acc[s] = __builtin_amdgcn_wmma_f32_16x16x4_f32(false, a, false, bv, (short)0,
                                                           acc[s], false, false);
        }
    }

    const int rhi = (lane >> 4) << 3;  // lanes 16-31 hold rows r+8
#pragma unroll
    for (int r = 0; r < 8; ++r) {
        const int coS = co0 + r + rhi;
        if (coS < Co) {
            const float bv = bias ? bias[coS] : 0.0f;
#pragma unroll
            for (int s = 0; s < 4; ++s) {
                const int n = nl + s * 16;
                float v = acc[s][r] + bv;
                if (Ab) v += Ab[(size_t)coS * cols + n];
                if (act) v = lrelu_f(v);
                Yb[(size_t)coS * cols + n] = v;
            }
        }
    }
}

// ---------------- max over last dim (cols) of (B*C, cols) -> (B*C,) ---------
__global__ void maxpool_m(const float* __restrict__ in, float* __restrict__ out, int cols)
{
    __shared__ float red[256];
    const int bc = blockIdx.x;
    const float* p = in + (size_t)bc * cols;
    float mx = -3.4e38f;
    for (int i = threadIdx.x; i < cols; i += blockDim.x) mx = fmaxf(mx, p[i]);
    red[threadIdx.x] = mx;
    __syncthreads();
    for (int s = 128; s > 0; s >>= 1) {
        if ((int)threadIdx.x < s) red[threadIdx.x] = fmaxf(red[threadIdx.x], red[threadIdx.x + s]);
        __syncthreads();
    }
    if (threadIdx.x == 0) out[bc] = red[0];
}

// ---------------- concat [pos_k(3) | x1max bcast(128) | glf(640)] -----------
__global__ void concat_stn(const float* __restrict__ posk, const float* __restrict__ x1max,
                           const float* __restrict__ glf, float* __restrict__ cat)
{
    size_t t = (size_t)blockIdx.x * blockDim.x + threadIdx.x;
    const size_t total = (size_t)BN * CI771 * MP;
    if (t >= total) return;
    const int m = (int)(t % MP);
    const int r = (int)(t / MP);
    const int c = r % CI771;
    const int b = r / CI771;
    float v;
    if (c < 3)        v = posk[((size_t)b * 3 + c) * MP + m];
    else if (c < 131) v = x1max[b * 128 + (c - 3)];
    else              v = glf[((size_t)b * 640 + (c - 131)) * MP + m];
    cat[t] = v;
}

// ---- concat [mis_pos(3) | pooled bcast(128) | f_global(512) | feat(128)] ---
__global__ void concat_pd(const float* __restrict__ mis, const float* __restrict__ pooled,
                          const float* __restrict__ fglob, const float* __restrict__ feat,
                          float* __restrict__ cat)
{
    size_t t = (size_t)blockIdx.x * blockDim.x + threadIdx.x;
    const size_t total = (size_t)BN * CI771 * MP;
    if (t >= total) return;
    const int m = (int)(t % MP);
    const int r = (int)(t / MP);
    const int c = r % CI771;
    const int b = r / CI771;
    float v;
    if (c < 3)        v = mis[((size_t)b * 3 + c) * MP + m];
    else if (c < 131) v = pooled[b * 128 + (c - 3)];
    else if (c < 643) v = fglob[b * GFDC + (c - 131)];
    else              v = feat[((size_t)b * 128 + (c - 643)) * MP + m];
    cat[t] = v;
}

// ---------------- fc2 (64x128) + lrelu + fc3 (9x64) -> trans (B,9) ----------
__global__ void fc_head(const float* __restrict__ pooled,
                        const float* __restrict__ fc2w, const float* __restrict__ fc2b,
                        const float* __restrict__ fc3w, const float* __restrict__ fc3b,
                        float* __restrict__ trans)
{
    __shared__ float pl[128];
    __shared__ float h[64];
    const int b = blockIdx.x;
    const int o = threadIdx.x;  // 64 threads
    pl[o]      = pooled[b * 128 + o];
    pl[o + 64] = pooled[b * 128 + o + 64];
    __syncthreads();
    float s = fc2b[o];
    for (int i = 0; i < 128; ++i) s += fc2w[o * 128 + i] * pl[i];
    h[o] = lrelu_f(s);
    __syncthreads();
    if (o < 9) {
        float s2 = fc3b[o];
        for (int i = 0; i < 64; ++i) s2 += fc3w[o * 64 + i] * h[i];
        trans[b * 9 + o] = s2;
    }
}

// ---------------- mis_pos[b,j,m] = sum_i pos_k[b,i,m] * T[b,i,j] ------------
__global__ void mispos_k(const float* __restrict__ posk, const float* __restrict__ trans,
                         float* __restrict__ mis)
{
    const int t = blockIdx.x * blockDim.x + threadIdx.x;
    if (t >= BN * MP) return;
    const int m = t % MP;
    const int b = t / MP;
    const float x = posk[((size_t)b * 3 + 0) * MP + m];
    const float y = posk[((size_t)b * 3 + 1) * MP + m];
    const float z = posk[((size_t)b * 3 + 2) * MP + m];
    const float* T = trans + b * 9;
    mis[((size_t)b * 3 + 0) * MP + m] = x * T[0] + y * T[3] + z * T[6];
    mis[((size_t)b * 3 + 1) * MP + m] = x * T[1] + y * T[4] + z * T[7];
    mis[((size_t)b * 3 + 2) * MP + m] = x * T[2] + y * T[5] + z * T[8];
}

// ---------------- KNN: top-32 nearest of N=8192 per query (set, unordered) --
#define KTILE 256
__global__ void knn_kernel(const float* __restrict__ pos, const float* __restrict__ mis,
                           int* __restrict__ idx)
{
    __shared__ float sx[KTILE], sy[KTILE], sz[KTILE];
    const int b = blockIdx.y;
    const int m = blockIdx.x * blockDim.x + threadIdx.x;  // blockDim = 128, MP % 128 == 0
    const float* pb = pos + (size_t)b * 3 * NP;
    const float qx = mis[((size_t)b * 3 + 0) * MP + m];
    const float qy = mis[((size_t)b * 3 + 1) * MP + m];
    const float qz = mis[((size_t)b * 3 + 2) * MP + m];
    float bd[KNN];
    int   bi[KNN];
#pragma unroll
    for (int j = 0; j < KNN; ++j) { bd[j] = 3.4e38f; bi[j] = 0; }
    float wv = 3.4e38f;  // current worst (max) of kept set
    int   wj = 0;
    for (int n0 = 0; n0 < NP; n0 += KTILE) {
        __syncthreads();
        for (int tt = threadIdx.x; tt < KTILE; tt += blockDim.x) {
            sx[tt] = pb[n0 + tt];
            sy[tt] = pb[NP + n0 + tt];
            sz[tt] = pb[2 * NP + n0 + tt];
        }
        // pull next tile toward the WGP while we scan this one
        if (n0 + KTILE < NP) {
            __builtin_prefetch(pb + n0 + KTILE + threadIdx.x, 0, 1);
            __builtin_prefetch(pb + NP + n0 + KTILE + threadIdx.x, 0, 1);
            __builtin_prefetch(pb + 2 * NP + n0 + KTILE + threadIdx.x, 0, 1);
        }
        __syncthreads();
        for (int p = 0; p < KTILE; ++p) {
            const float dx = sx[p] - qx, dy = sy[p] - qy, dz = sz[p] - qz;
            const float d2 = dx * dx + dy * dy + dz * dz;
            if (d2 < wv) {
                bd[wj] = d2; bi[wj] = n0 + p;
                wv = bd[0]; wj = 0;
#pragma unroll
                for (int j = 1; j < KNN; ++j) if (bd[j] > wv) { wv = bd[j]; wj = j; }
            }
        }
    }
    for (int j = 0; j < KNN; ++j) idx[((size_t)b * MP + m) * KNN + j] = bi[j];
}

// ---------------- build 10-channel KNN feature, layout (B, 10, MC*KNN) ------
// channels: [dist, nbr-ctr(3), ctr(3), nbr(3)], column index = mm*KNN + k
__global__ void feat_build(const float* __restrict__ pos, const float* __restrict__ mis,
                           const int* __restrict__ idx, float* __restrict__ f10, int m0)
{
    const int t = blockIdx.x * blockDim.x + threadIdx.x;
    if (t >= BN * MC * KNN) return;
    const int k  = t % KNN;
    const int r  = t / KNN;
    const int mm = r % MC;
    const int b  = r / MC;
    const int m  = m0 + mm;
    const int n  = idx[((size_t)b * MP + m) * KNN + k];
    const float cx = mis[((size_t)b * 3 + 0) * MP + m];
    const float cy = mis[((size_t)b * 3 + 1) * MP + m];
    const float cz = mis[((size_t)b * 3 + 2) * MP + m];
    const float nx = pos[((size_t)b * 3 + 0) * NP + n];
    const float ny = pos[((size_t)b * 3 + 1) * NP + n];
    const float nz = pos[((size_t)b * 3 + 2) * NP + n];
    const float dx = nx - cx, dy = ny - cy, dz = nz - cz;
    const float d2 = dx * dx + dy * dy + dz * dz;
    const float dist = sqrtf(fmaxf(d2, 1e-12f));
    const size_t base = (size_t)b * 10 * CCOLS + (size_t)mm * KNN + k;
    f10[base + 0 * (size_t)CCOLS] = dist;
    f10[base + 1 * (size_t)CCOLS] = dx;
    f10[base + 2 * (size_t)CCOLS] = dy;
    f10[base + 3 * (size_t)CCOLS] = dz;
    f10[base + 4 * (size_t)CCOLS] = cx;
    f10[base + 5 * (size_t)CCOLS] = cy;
    f10[base + 6 * (size_t)CCOLS] = cz;
    f10[base + 7 * (size_t)CCOLS] = nx;
    f10[base + 8 * (size_t)CCOLS] = ny;
    f10[base + 9 * (size_t)CCOLS] = nz;
}

// ---- cat128[0:64] = x ; cat128[64:128] = broadcast(max over k of x) --------
__global__ void maxk_concat(const float* __restrict__ x, float* __restrict__ cat)
{
    const int t = blockIdx.x * blockDim.x + threadIdx.x;
    if (t >= BN * 64 * MC) return;
    const int mm = t % MC;
    const int r  = t / MC;
    const int c  = r % 64;
    const int b  = r / 64;
    const size_t ib = (size_t)b * 64 * CCOLS + (size_t)c * CCOLS + (size_t)mm * KNN;
    const size_t o1 = (size_t)b * 128 * CCOLS + (size_t)c * CCOLS + (size_t)mm * KNN;
    const size_t o2 = o1 + (size_t)64 * CCOLS;
    float mx = -3.4e38f;
#pragma unroll
    for (int k = 0; k < KNN; ++k) { const float v = x[ib + k]; cat[o1 + k] = v; mx = fmaxf(mx, v); }
#pragma unroll
    for (int k = 0; k < KNN; ++k) cat[o2 + k] = mx;
}

// ---------------- mis_feat[b,c,m0+mm] = max over k --------------------------
__global__ void maxk_out(const float* __restrict__ x, float* __restrict__ outFeat, int m0)
{
    const int t = blockIdx.x * blockDim.x + threadIdx.x;
    if (t >= BN * 128 * MC) return;
    const int mm = t % MC;
    const int r  = t / MC;
    const int c  = r % 128;
    const int b  = r / 128;
    const size_t ib = (size_t)b * 128 * CCOLS + (size_t)c * CCOLS + (size_t)mm * KNN;
    float mx = -3.4e38f;
#pragma unroll
    for (int k = 0; k < KNN; ++k) mx = fmaxf(mx, x[ib + k]);
    outFeat[((size_t)b * 128 + c) * MP + m0 + mm] = mx;
}

// ---------------- out_pos = mis_base + displacement -------------------------
__global__ void addpos(const float* __restrict__ a, const float* __restrict__ d,
                       float* __restrict__ o)
{
    const int t = blockIdx.x * blockDim.x + threadIdx.x;
    if (t >= BN * 3 * MP) return;
    o[t] = a[t] + d[t];
}

// ---------------------------------------------------------------------------
// Input index map. Assumes the harness flattens the inputs dict in order
// (pos, pos_k, local_feat_k, global_local_feats_k, f_global, params) and the
// params pytree via jax tree order: dict keys sorted alphabetically at every
// level; each _conv {"w","b"} flattens as (b, w).
// params: mlp2{conv1,conv2}, pos_offset{conv1,conv2,conv3}, stn3d{conv1,conv2,fc2,fc3}
// each _res {c1, c2a, c2b, short}.
// ---------------------------------------------------------------------------
enum {
    IN_POS = 0, IN_POSK, IN_LOCALFEAT, IN_GLF, IN_FGLOBAL,
    M2C1_C1B, M2C1_C1W, M2C1_C2AB, M2C1_C2AW, M2C1_C2BB, M2C1_C2BW, M2C1_SHB, M2C1_SHW,
    M2C2_C1B, M2C2_C1W, M2C2_C2AB, M2C2_C2AW, M2C2_C2BB, M2C2_C2BW, M2C2_SHB, M2C2_SHW,
    PO1_C1B, PO1_C1W, PO1_C2AB, PO1_C2AW, PO1_C2BB, PO1_C2BW, PO1_SHB, PO1_SHW,
    PO2_C1B, PO2_C1W, PO2_C2AB, PO2_C2AW, PO2_C2BB, PO2_C2BW, PO2_SHB, PO2_SHW,
    PO3_B, PO3_W,
    ST1_C1B, ST1_C1W, ST1_C2AB, ST1_C2AW, ST1_C2BB, ST1_C2BW, ST1_SHB, ST1_SHW,
    ST2_C1B, ST2_C1W, ST2_C2AB, ST2_C2AW, ST2_C2BB, ST2_C2BW, ST2_SHB, ST2_SHW,
    FC2_B, FC2_W, FC3_B, FC3_W
};

extern "C" void kernel_launch(void* const* d_in, const int* in_sizes, int n_in,
                              void* d_out, int out_size, void* d_ws, size_t ws_size,
                              hipStream_t stream)
{
    (void)in_sizes; (void)n_in; (void)out_size; (void)ws_size;
#define PW(i) ((const float*)d_in[(i)])
    const float* pos   = PW(IN_POS);
    const float* posk  = PW(IN_POSK);
    const float* glf   = PW(IN_GLF);
    const float* fglob = PW(IN_FGLOBAL);

    float* out     = (float*)d_out;
    float* outPos  = out;                       // (B,3,M)
    float* outFeat = out + (size_t)BN * 3 * MP; // (B,128,M)

    // ---- workspace bump allocator (floats) ----
    float* w = (float*)d_ws;
    size_t off = 0;
    auto A = [&](size_t n) { float* p = w + off; off += n; return p; };
    float* misbase = A((size_t)BN * 3 * MP);
    float* disp    = A((size_t)BN * 3 * MP);
    float* trans   = A((size_t)BN * 9);
    float* x1max   = A((size_t)BN * 128);
    float* pooled  = A((size_t)BN * 128);
    float* pooled2 = A((size_t)BN * 128);
    float* cat771  = A((size_t)BN * CI771 * MP);
    float* t64M    = A((size_t)BN * 64 * MP);
    float* t128a   = A((size_t)BN * 128 * MP);
    float* t128b   = A((size_t)BN * 128 * MP);
    float* t128s   = A((size_t)BN * 128 * MP);
    float* h128    = A((size_t)BN * 128 * MP);
    int*   knnidx  = (int*)A((size_t)BN * MP * KNN);
    float* f10     = A((size_t)BN * 10 * CCOLS);
    float* u64a    = A((size_t)BN * 64 * CCOLS);
    float* u64b    = A((size_t)BN * 64 * CCOLS);
    float* u64s    = A((size_t)BN * 64 * CCOLS);
    float* u64o    = A((size_t)BN * 64 * CCOLS);
    float* cat128  = A((size_t)BN * 128 * CCOLS);
    float* v128a   = A((size_t)BN * 128 * CCOLS);
    float* v128s   = A((size_t)BN * 128 * CCOLS);
    float* v128o   = A((size_t)BN * 128 * CCOLS);

    auto gemm = [&](const float* Wt, const float* Bv, const float* X, float* Y,
                    const float* add, int Co, int Ci, int cols, int act) {
        const int tiles = (cols >> 6) * ((Co + 15) >> 4);
        dim3 g((tiles + 7) / 8, BN);
        gemm_wmma_f32<<<g, 256, 0, stream>>>(Wt, Bv, X, Y, add, Co, Ci, cols, act);
    };

    // ===== Stage 1: stn3d.conv1 mlp_res on pos_k (3 -> 64 -> 128) ==========
    gemm(PW(ST1_C1W),  PW(ST1_C1B),  posk,  t64M,  nullptr, 64,  3,   MP, 1);
    gemm(PW(ST1_C2AW), PW(ST1_C2AB), t64M,  t128a, nullptr, 128, 64,  MP, 1);
    gemm(PW(ST1_SHW),  PW(ST1_SHB),  posk,  t128s, nullptr, 128, 3,   MP, 0);
    gemm(PW(ST1_C2BW), PW(ST1_C2BB), t128a, t128b, t128s,   128, 128, MP, 0);
    maxpool_m<<<BN * 128, 256, 0, stream>>>(t128b, x1max, MP);

    // ===== Stage 2: concat + stn3d.conv2 (771 -> 128 -> 128) + head ========
    {
        size_t tot = (size_t)BN * CI771 * MP;
        concat_stn<<<(unsigned)((tot + 255) / 256), 256, 0, stream>>>(posk, x1max, glf, cat771);
    }
    gemm(PW(ST2_C1W),  PW(ST2_C1B),  cat771, t128a, nullptr, 128, CI771, MP, 1);
    gemm(PW(ST2_C2AW), PW(ST2_C2AB), t128a,  t128b, nullptr, 128, 128,   MP, 1);
    gemm(PW(ST2_SHW),  PW(ST2_SHB),  cat771, t128s, nullptr, 128, CI771, MP, 0);
    gemm(PW(ST2_C2BW), PW(ST2_C2BB), t128b,  h128,  t128s,   128, 128,   MP, 0);
    maxpool_m<<<BN * 128, 256, 0, stream>>>(h128, pooled, MP);
    fc_head<<<BN, 64, 0, stream>>>(pooled, PW(FC2_W), PW(FC2_B), PW(FC3_W), PW(FC3_B), trans);

    // ===== Stage 3: mis_pos = pos_k^T x trans3d ============================
    mispos_k<<<(BN * MP + 255) / 256, 256, 0, stream>>>(posk, trans, misbase);

    // ===== Stage 4: KNN indices ============================================
    knn_kernel<<<dim3(MP / 128, BN), 128, 0, stream>>>(pos, misbase, knnidx);

    // ===== Stage 5: point_mis_feat, chunked over M =========================
    for (int m0 = 0; m0 < MP; m0 += MC) {
        feat_build<<<(BN * MC * KNN + 255) / 256, 256, 0, stream>>>(pos, misbase, knnidx, f10, m0);
        // conv_res1: _res(10, 64, 64)
        gemm(PW(M2C1_C1W),  PW(M2C1_C1B),  f10,  u64a, nullptr, 64, 10, CCOLS, 1);
        gemm(PW(M2C1_C2AW), PW(M2C1_C2AB), u64a, u64b, nullptr, 64, 64, CCOLS, 1);
        gemm(PW(M2C1_SHW),  PW(M2C1_SHB),  f10,  u64s, nullptr, 64, 10, CCOLS, 0);
        gemm(PW(M2C1_C2BW), PW(M2C1_C2BB), u64b, u64o, u64s,    64, 64, CCOLS, 0);
        maxk_concat<<<(BN * 64 * MC + 255) / 256, 256, 0, stream>>>(u64o, cat128);
        // conv_res2: _res(128, 64, 128)
        gemm(PW(M2C2_C1W),  PW(M2C2_C1B),  cat128, u64a,  nullptr, 64,  128, CCOLS, 1);
        gemm(PW(M2C2_C2AW), PW(M2C2_C2AB), u64a,   v128a, nullptr, 128, 64,  CCOLS, 1);
        gemm(PW(M2C2_SHW),  PW(M2C2_SHB),  cat128, v128s, nullptr, 128, 128, CCOLS, 0);
        gemm(PW(M2C2_C2BW), PW(M2C2_C2BB), v128a,  v128o, v128s,   128, 128, CCOLS, 0);
        maxk_out<<<(BN * 128 * MC + 255) / 256, 256, 0, stream>>>(v128o, outFeat, m0);
    }

    // ===== Stage 6: pos_offset.conv1 mlp_res on mis_pos (3 -> 64 -> 128) ===
    gemm(PW(PO1_C1W),  PW(PO1_C1B),  misbase, t64M,  nullptr, 64,  3,   MP, 1);
    gemm(PW(PO1_C2AW), PW(PO1_C2AB), t64M,    t128a, nullptr, 128, 64,  MP, 1);
    gemm(PW(PO1_SHW),  PW(PO1_SHB),  misbase, t128s, nullptr, 128, 3,   MP, 0);
    gemm(PW(PO1_C2BW), PW(PO1_C2BB), t128a,   h128,  t128s,   128, 128, MP, 0);
    maxpool_m<<<BN * 128, 256, 0, stream>>>(h128, pooled2, MP);

    // ===== Stage 7: concat [mis_pos | pooled2 | f_global | mis_feat] =======
    {
        size_t tot = (size_t)BN * CI771 * MP;
        concat_pd<<<(unsigned)((tot + 255) / 256), 256, 0, stream>>>(misbase, pooled2, fglob, outFeat, cat771);
    }

    // ===== Stage 8: pos_offset.conv2 _res(771,128,64), final lrelu =========
    gemm(PW(PO2_C1W),  PW(PO2_C1B),  cat771, t128a, nullptr, 128, CI771, MP, 1);
    gemm(PW(PO2_C2AW), PW(PO2_C2AB), t128a,  t64M,  nullptr, 64,  128,   MP, 1);
    gemm(PW(PO2_SHW),  PW(PO2_SHB),  cat771, u64s,  nullptr, 64,  CI771, MP, 0);
    gemm(PW(PO2_C2BW), PW(PO2_C2BB), t64M,   u64o,  u64s,    64,  64,    MP, 1);

    // ===== Stage 9: conv3 (3 x 64) -> displacement =========================
    gemm(PW(PO3_W), PW(PO3_B), u64o, disp, nullptr, 3, 64, MP, 0);

    // ===== Stage 10: final mis_pos = base + displacement ===================
    addpos<<<(BN * 3 * MP + 255) / 256, 256, 0, stream>>>(misbase, disp, outPos);
#undef PW
}